// LiftedStructureLoss_57629871178314
// MI455X (gfx1250) — compile-verified
//
#include <hip/hip_runtime.h>

#define NROW 4096
#define DIM  512
#define TILE 16
#define NT   (NROW / TILE)   // 256 tiles per axis
#define WAVES 8
#define LDSTR (DIM + 4)      // pad: row stride 516 floats -> conflict-free ds_load_b64,
                             // and 516*4 B is 16B-aligned for b128 staging stores
#define MARGIN 1.0f

typedef float v2f __attribute__((ext_vector_type(2)));
typedef float v4f __attribute__((ext_vector_type(4)));
typedef float v8f __attribute__((ext_vector_type(8)));

#define WMMA_F32(a, b, c) \
    __builtin_amdgcn_wmma_f32_16x16x4_f32(false, (a), false, (b), (short)0, (c), false, false)

// ---------------------------------------------------------------- mag = ||x||^2
__global__ void mag_kernel(const float* __restrict__ X, float* __restrict__ mag) {
    __shared__ float red[128];
    const int row = blockIdx.x;
    const int t   = threadIdx.x;
    // 512 floats = 128 float4s: exactly one b128 load per thread
    const v4f v = *(const v4f*)(X + (size_t)row * DIM + 4 * t);
    float s = v.x * v.x + v.y * v.y + v.z * v.z + v.w * v.w;
    red[t] = s;
    __syncthreads();
    for (int off = 64; off > 0; off >>= 1) {
        if (t < off) red[t] += red[t + off];
        __syncthreads();
    }
    if (t == 0) mag[row] = red[0];
}

// ------------------------------------- stage 16xDIM i-strip into LDS (b128 ops)
__device__ __forceinline__ void load_strip(const float* __restrict__ X, int i_base,
                                           float (*As)[LDSTR]) {
    const int NV = TILE * DIM / 4;                     // 2048 float4s
    for (int li = threadIdx.x; li < NV; li += blockDim.x) {
        const int r = li >> 7;                         // / (DIM/4)
        const int c = (li & (DIM / 4 - 1)) << 2;       // float4 column
        *(v4f*)(&As[r][c]) = *(const v4f*)(X + (size_t)(i_base + r) * DIM + c);
    }
}

// -------- NJ consecutive 16x16 tiles of X · X^T via V_WMMA_F32_16X16X4_F32.
// A-frag (ISA 7.12.2, 32-bit A 16x4): lane l holds A[l%16][2*(l>>4) + {0,1}].
// B is the transpose-symmetric operand: same addressing against the j-strip.
// One ds_load A-fragment feeds NJ WMMAs into NJ independent accumulator chains.
template <int NJ>
__device__ __forceinline__ void sim_tileN(const float* __restrict__ X,
                                          const float (*As)[LDSTR],
                                          int jb, int lane, v8f c[NJ]) {
    const int m  = lane & 15;
    const int kk = (lane >> 4) << 1;
    const float* B[NJ];
    #pragma unroll
    for (int q = 0; q < NJ; q++)
        B[q] = X + (size_t)(jb + q * TILE + m) * DIM + kk;
    const float* Ar = &As[m][kk];
    #pragma unroll
    for (int q = 0; q < NJ; q++) c[q] = {};
    #pragma unroll 2
    for (int k0 = 0; k0 < DIM; k0 += 4) {
        const v2f a = *(const v2f*)(Ar + k0);
        #pragma unroll
        for (int q = 0; q < NJ; q++) {
            const v2f b = *(const v2f*)(B[q] + k0);
            c[q] = WMMA_F32(a, b, c[q]);
        }
    }
}

// C/D layout: VGPR r, lanes 0-15 -> M=r, lanes 16-31 -> M=r+8; N = lane%16.
__device__ __forceinline__ int row_of(int lane, int r) {
    return (lane < 16) ? r : r + 8;
}

// ----------------------------- pass 1: row_neg[i] = sum_j(neg) exp(1 - dist_ij)
__global__ void rowneg_kernel(const float* __restrict__ X, const int* __restrict__ tgt,
                              const float* __restrict__ mag, float* __restrict__ row_neg) {
    __shared__ float As[TILE][LDSTR];
    __shared__ float part[WAVES][32][8];
    __shared__ float wsum[WAVES][TILE];

    const int i_base = blockIdx.x * TILE;
    const int t    = threadIdx.x;
    const int lane = t & 31;
    const int wave = t >> 5;
    const int m    = lane & 15;

    load_strip(X, i_base, As);
    __syncthreads();

    float mi[8]; int ti[8];
    #pragma unroll
    for (int r = 0; r < 8; r++) {
        const int i = i_base + row_of(lane, r);
        mi[r] = mag[i];
        ti[r] = tgt[i];
    }

    float acc[8];
    #pragma unroll
    for (int r = 0; r < 8; r++) acc[r] = 0.0f;

    auto epi = [&](const v8f& c, int j_base) {
        const int   j  = j_base + m;                  // column this lane owns
        const float mj = mag[j];
        const int   tj = tgt[j];
        #pragma unroll
        for (int r = 0; r < 8; r++) {
            float d2   = fmaxf(mi[r] + mj - 2.0f * c[r], 0.0f);
            float dist = __builtin_amdgcn_sqrtf(d2);  // sqrt(0) == 0
            if ((ti[r] != tj) && (dist != 0.0f))
                acc[r] += __expf(MARGIN - dist);
        }
    };

    // NT = 256 = 8 iterations of (8 waves x 4 tiles): no remainder handling
    for (int jt = 4 * wave; jt < NT; jt += 4 * WAVES) {
        v8f c[4];
        sim_tileN<4>(X, As, jt * TILE, lane, c);
        #pragma unroll
        for (int q = 0; q < 4; q++) epi(c[q], (jt + q) * TILE);
    }

    #pragma unroll
    for (int r = 0; r < 8; r++) part[wave][lane][r] = acc[r];
    __syncthreads();

    // fixed-order reduction over the 16 lanes that share a row (deterministic)
    if (t < WAVES * TILE) {
        const int w = t / TILE, M = t % TILE;
        const int r  = (M < 8) ? M : M - 8;
        const int l0 = (M < 8) ? 0 : 16;
        float s = 0.0f;
        for (int ll = 0; ll < 16; ll++) s += part[w][l0 + ll][r];
        wsum[w][M] = s;
    }
    __syncthreads();
    if (t < TILE) {
        float s = 0.0f;
        for (int w = 0; w < WAVES; w++) s += wsum[w][t];
        row_neg[i_base + t] = s;
    }
}

// -------- pass 2: loss over upper-triangle positive pairs + positive-pair count
__global__ void loss_kernel(const float* __restrict__ X, const int* __restrict__ tgt,
                            const float* __restrict__ mag, const float* __restrict__ row_neg,
                            float* __restrict__ loss_part, float* __restrict__ cnt_part) {
    __shared__ float As[TILE][LDSTR];
    __shared__ float redl[256];
    __shared__ float redc[256];

    const int i_tile = blockIdx.x;
    const int i_base = i_tile * TILE;
    const int t    = threadIdx.x;
    const int lane = t & 31;
    const int wave = t >> 5;
    const int m    = lane & 15;

    load_strip(X, i_base, As);
    __syncthreads();

    float mi[8], rni[8]; int ti[8], irow[8];
    #pragma unroll
    for (int r = 0; r < 8; r++) {
        const int i = i_base + row_of(lane, r);
        irow[r] = i;
        mi[r]  = mag[i];
        ti[r]  = tgt[i];
        rni[r] = row_neg[i];
    }

    float lacc = 0.0f, cacc = 0.0f;

    auto epi = [&](const v8f& c, int j_base) {
        const int   j   = j_base + m;
        const float mj  = mag[j];
        const int   tj  = tgt[j];
        const float rnj = row_neg[j];
        #pragma unroll
        for (int r = 0; r < 8; r++) {
            float d2   = fmaxf(mi[r] + mj - 2.0f * c[r], 0.0f);
            float dist = __builtin_amdgcn_sqrtf(d2);
            if ((ti[r] == tj) && (irow[r] < j)) {          // upper-tri positive pair
                float J = __logf(rni[r] + rnj) + dist;
                J = fmaxf(J, 0.0f);
                lacc = fmaf(J, J, lacc);
                cacc += 1.0f;                              // len_p = 2 * sum(cacc)
            }
        }
    };

    // upper-triangle tiles, 4 per wave-iteration; wave-uniform remainder variants
    for (int jt = i_tile + 4 * wave; jt < NT; jt += 4 * WAVES) {
        const int rem = NT - jt;
        if (rem >= 4) {
            v8f c[4];
            sim_tileN<4>(X, As, jt * TILE, lane, c);
            #pragma unroll
            for (int q = 0; q < 4; q++) epi(c[q], (jt + q) * TILE);
        } else if (rem == 3) {
            v8f c[3];
            sim_tileN<3>(X, As, jt * TILE, lane, c);
            #pragma unroll
            for (int q = 0; q < 3; q++) epi(c[q], (jt + q) * TILE);
        } else if (rem == 2) {
            v8f c[2];
            sim_tileN<2>(X, As, jt * TILE, lane, c);
            #pragma unroll
            for (int q = 0; q < 2; q++) epi(c[q], (jt + q) * TILE);
        } else {
            v8f c[1];
            sim_tileN<1>(X, As, jt * TILE, lane, c);
            epi(c[0], jt * TILE);
        }
    }

    redl[t] = lacc;
    redc[t] = cacc;
    __syncthreads();
    for (int off = 128; off > 0; off >>= 1) {
        if (t < off) { redl[t] += redl[t + off]; redc[t] += redc[t + off]; }
        __syncthreads();
    }
    if (t == 0) { loss_part[blockIdx.x] = redl[0]; cnt_part[blockIdx.x] = redc[0]; }
}

// ------------------------------------------- final: sum 256 partials, normalize
__global__ void final_kernel(const float* __restrict__ lp, const float* __restrict__ cp,
                             float* __restrict__ out) {
    __shared__ float rl[256];
    __shared__ float rc[256];
    const int t = threadIdx.x;
    rl[t] = lp[t];      // NT == 256 == blockDim.x
    rc[t] = cp[t];
    __syncthreads();
    for (int off = 128; off > 0; off >>= 1) {
        if (t < off) { rl[t] += rl[t + off]; rc[t] += rc[t + off]; }
        __syncthreads();
    }
    if (t == 0) out[0] = rl[0] / (2.0f * rc[0]);   // divide by symmetric positive count
}

extern "C" void kernel_launch(void* const* d_in, const int* in_sizes, int n_in,
                              void* d_out, int out_size, void* d_ws, size_t ws_size,
                              hipStream_t stream) {
    const float* X   = (const float*)d_in[0];   // [4096, 512] f32
    const int*   tgt = (const int*)d_in[1];     // [4096] i32
    float* ws        = (float*)d_ws;
    float* mag       = ws;                      // 4096
    float* row_neg   = ws + NROW;               // 4096
    float* loss_part = ws + 2 * NROW;           // 256
    float* cnt_part  = ws + 2 * NROW + NT;      // 256
    float* out       = (float*)d_out;

    mag_kernel   <<<NROW, 128, 0, stream>>>(X, mag);
    rowneg_kernel<<<NT,   256, 0, stream>>>(X, tgt, mag, row_neg);
    loss_kernel  <<<NT,   256, 0, stream>>>(X, tgt, mag, row_neg, loss_part, cnt_part);
    final_kernel <<<1,    256, 0, stream>>>(loss_part, cnt_part, out);
}